// PITWrapper_17059610100257
// MI455X (gfx1250) — compile-verified
//
#include <hip/hip_runtime.h>
#include <math.h>

// ---------------------------------------------------------------------------
// PIT SI-SDR loss, B=32, C=3, T=96000, fp32.
//
// Memory-bound: 74 MB of input @ 23.3 TB/s => ~3.2 us floor. One streaming
// pass computes per batch: raw Gram dot[3][3] (est_i . ref_j) via
// V_WMMA_F32_16X16X4_F32 (5 batches * 3 channels packed in the 16x16 tile,
// block-diagonal results), plus sum / sumsq of every row via packed VALU,
// all fed from the same b128 loads. Mean-centering folded in at finalize.
//
// K-slot remap trick: dot products are order-invariant, so each lane loads
// one float4 at (t + 4*half) covering TWO WMMA K=4 steps:
//   WMMA0 uses floats {0,1} of each half  -> samples {t,t+1,t+4,t+5}
//   WMMA1 uses floats {2,3} of each half  -> samples {t+2,t+3,t+6,t+7}
// A and B use the same assignment, so partial dots are exact.
// ---------------------------------------------------------------------------

typedef __attribute__((ext_vector_type(2))) float v2f;
typedef __attribute__((ext_vector_type(4))) float v4f;
typedef __attribute__((ext_vector_type(8))) float v8f;

#define B_  32
#define C_  3
#define T_  96000
#define NBG 7                      // ceil(32 batches / 5 per 16-row tile)
#define NCHUNK 200                 // T sub-chunks per batch-group
#define CHUNKLEN (T_ / NCHUNK)     // 480, multiple of 16 (60 x 8-sample steps)
#define WAVES_PER_BLOCK 8
#define EPSF 1e-8f

// scratch layout (floats)
#define WS_DOT 0                   // B*9  raw dot products
#define WS_SE  288                 // 96   sum(est row)
#define WS_SQE 384                 // 96   sum(est^2)
#define WS_SR  480                 // 96   sum(ref row)
#define WS_SQR 576                 // 96   sum(ref^2)
#define WS_TOT 672

__global__ void zero_ws_kernel(float* __restrict__ ws) {
  int i = blockIdx.x * blockDim.x + threadIdx.x;
  if (i < WS_TOT) ws[i] = 0.0f;
}

__global__ __launch_bounds__(256)
void gram_wmma_kernel(const float* __restrict__ est,
                      const float* __restrict__ ref,
                      float* __restrict__ ws) {
  const int lane = threadIdx.x & 31;
  const int wave = threadIdx.x >> 5;
  const int half = lane >> 4;      // K-parity half of the wave
  const int r    = lane & 15;      // M row (A) == N col (B) index 0..15

  const int bg    = blockIdx.y;                          // batch group (5 batches)
  const int chunk = blockIdx.x * WAVES_PER_BLOCK + wave; // T chunk
  const int t0 = chunk * CHUNKLEN;
  const int t1 = t0 + CHUNKLEN;

  // flat row = batch*3 + channel = bg*15 + r  (rows 15.. clamp to a live row
  // so the dummy lanes hit cache instead of adding DRAM traffic)
  int flatRow = bg * 15 + r;
  const bool rowValid = (r < 15) && (flatRow < B_ * C_);
  if (!rowValid) flatRow = bg * 15;

  const float* ep = est + (size_t)flatRow * T_;
  const float* rp = ref + (size_t)flatRow * T_;
  const int koff = 4 * half;       // half 0 -> t..t+3, half 1 -> t+4..t+7

  v8f acc = {};                    // 16x16 f32 C/D accumulator (8 VGPRs)
  float se = 0.f, sqe = 0.f, sr = 0.f, sqr = 0.f;

  #pragma unroll 2
  for (int t = t0; t < t1; t += 8) {
    v4f a = *(const v4f*)(ep + t + koff);   // global_load_b128, 16B aligned
    v4f b = *(const v4f*)(rp + t + koff);

    v2f a0 = {a[0], a[1]}, a1 = {a[2], a[3]};
    v2f b0 = {b[0], b[1]}, b1 = {b[2], b[3]};

    acc = __builtin_amdgcn_wmma_f32_16x16x4_f32(false, a0, false, b0,
                                                (short)0, acc, false, false);
    acc = __builtin_amdgcn_wmma_f32_16x16x4_f32(false, a1, false, b1,
                                                (short)0, acc, false, false);

    se  += (a[0] + a[1]) + (a[2] + a[3]);
    sqe += (a[0] * a[0] + a[1] * a[1]) + (a[2] * a[2] + a[3] * a[3]);
    sr  += (b[0] + b[1]) + (b[2] + b[3]);
    sqr += (b[0] * b[0] + b[1] * b[1]) + (b[2] * b[2] + b[3] * b[3]);
  }

  // fold the two half-wave partials of each row together
  se  += __shfl_xor(se,  16, 32);
  sqe += __shfl_xor(sqe, 16, 32);
  sr  += __shfl_xor(sr,  16, 32);
  sqr += __shfl_xor(sqr, 16, 32);

  if (rowValid && half == 0) {
    atomicAdd(&ws[WS_SE  + flatRow], se);
    atomicAdd(&ws[WS_SQE + flatRow], sqe);
    atomicAdd(&ws[WS_SR  + flatRow], sr);
    atomicAdd(&ws[WS_SQR + flatRow], sqr);
  }

  // D layout: lane (half, r) holds D[M = v + 8*half][N = r] in acc[v].
  // Keep only block-diagonal (same-batch) 3x3 entries.
  #pragma unroll
  for (int v = 0; v < 8; ++v) {
    const int M = v + 8 * half;
    if (M < 15 && r < 15 && (M / 3) == (r / 3)) {
      const int b = bg * 5 + M / 3;
      if (b < B_) {
        const int i = M % 3, j = r % 3;
        atomicAdd(&ws[WS_DOT + (b * 3 + i) * 3 + j], acc[v]);
      }
    }
  }
}

__global__ void finalize_kernel(const float* __restrict__ ws,
                                float* __restrict__ out) {
  const int b = threadIdx.x;       // one batch per lane, 32 lanes
  const float invT = 1.0f / (float)T_;

  float d[3][3], se[3], sqe[3], sr[3], sqr[3];
  #pragma unroll
  for (int i = 0; i < 3; ++i) {
    se[i]  = ws[WS_SE  + b * 3 + i];
    sqe[i] = ws[WS_SQE + b * 3 + i];
    sr[i]  = ws[WS_SR  + b * 3 + i];
    sqr[i] = ws[WS_SQR + b * 3 + i];
    #pragma unroll
    for (int j = 0; j < 3; ++j) d[i][j] = ws[WS_DOT + (b * 3 + i) * 3 + j];
  }

  float ee[3], re[3], pair[3][3];
  #pragma unroll
  for (int i = 0; i < 3; ++i) ee[i] = sqe[i] - se[i] * se[i] * invT;
  #pragma unroll
  for (int j = 0; j < 3; ++j) re[j] = sqr[j] - sr[j] * sr[j] * invT;

  #pragma unroll
  for (int i = 0; i < 3; ++i)
    #pragma unroll
    for (int j = 0; j < 3; ++j) {
      float dc = d[i][j] - se[i] * sr[j] * invT;     // centered <e_i, r_j>
      float te = dc * dc / (re[j] + EPSF);           // ||s_target||^2
      float ne = ee[i] - te;                         // ||e_noise||^2
      pair[i][j] = 10.0f * log10f(te / (ne + EPSF) + EPSF);
    }

  const int P[6][3] = {{0,1,2},{0,2,1},{1,0,2},{1,2,0},{2,0,1},{2,1,0}};
  float best = -INFINITY;
  #pragma unroll
  for (int p = 0; p < 6; ++p) {
    float s = (pair[0][P[p][0]] + pair[1][P[p][1]] + pair[2][P[p][2]]) *
              (1.0f / 3.0f);
    best = fmaxf(best, s);
  }

  // mean over the 32 batches (wave32 butterfly), negate
  float v = best;
  #pragma unroll
  for (int m = 16; m > 0; m >>= 1) v += __shfl_xor(v, m, 32);
  if (b == 0) out[0] = -v * (1.0f / 32.0f);
}

extern "C" void kernel_launch(void* const* d_in, const int* in_sizes, int n_in,
                              void* d_out, int out_size, void* d_ws,
                              size_t ws_size, hipStream_t stream) {
  (void)in_sizes; (void)n_in; (void)out_size; (void)ws_size;
  const float* est = (const float*)d_in[0];
  const float* ref = (const float*)d_in[1];
  float* out = (float*)d_out;
  float* ws  = (float*)d_ws;

  zero_ws_kernel<<<dim3((WS_TOT + 255) / 256), dim3(256), 0, stream>>>(ws);

  dim3 grid(NCHUNK / WAVES_PER_BLOCK, NBG);   // 25 x 7 blocks, 8 waves each
  gram_wmma_kernel<<<grid, dim3(256), 0, stream>>>(est, ref, ws);

  finalize_kernel<<<1, 32, 0, stream>>>(ws, out);
}